// MPNNEncoder_44504451121831
// MI455X (gfx1250) — compile-verified
//
#include <hip/hip_runtime.h>
#include <math.h>

#define NNODES 40960
#define NEDGES 655360
#define NGRAPH 128
#define PER    320
#define FIN    64
#define HID    128

typedef __attribute__((ext_vector_type(2))) float v2f;
typedef __attribute__((ext_vector_type(8))) float v8f;

// ---------------- utility: zero fills ----------------
__global__ void zero_i32(int* p, int n){ int i = blockIdx.x*blockDim.x+threadIdx.x; if(i<n) p[i]=0; }
__global__ void zero_f32(float* p, int n){ int i = blockIdx.x*blockDim.x+threadIdx.x; if(i<n) p[i]=0.f; }

// ---------------- degrees + symmetric norm ----------------
__global__ void degree_count(const int* __restrict__ src, const int* __restrict__ dst,
                             int* cnt_out, int* cnt_in){
  int e = blockIdx.x*blockDim.x+threadIdx.x;
  if(e<NEDGES){ atomicAdd(&cnt_out[src[e]],1); atomicAdd(&cnt_in[dst[e]],1); }
}

__global__ void norm_kernel(const int* __restrict__ cnt_out, const int* __restrict__ cnt_in,
                            float* norm_out, float* norm_in){
  int n = blockIdx.x*blockDim.x+threadIdx.x;
  if(n<NNODES){
    float a = (float)cnt_out[n]; if(a<1.f) a=1.f;
    float b = (float)cnt_in[n];  if(b<1.f) b=1.f;
    norm_out[n]=rsqrtf(a); norm_in[n]=rsqrtf(b);
  }
}

// ---------------- CSR build: exclusive scan (single block) + scatter ----------------
__global__ void scan_kernel(const int* __restrict__ cnt_in, int* __restrict__ rowptr){
  __shared__ int part[1024];
  const int CH = NNODES/1024;               // 40 exactly
  int t = threadIdx.x;
  int begin = t*CH;
  int s = 0;
  for(int i=0;i<CH;i++) s += cnt_in[begin+i];
  part[t]=s; __syncthreads();
  for(int off=1; off<1024; off<<=1){
    int v = (t>=off)?part[t-off]:0;
    __syncthreads();
    part[t]+=v;
    __syncthreads();
  }
  int run = (t==0)?0:part[t-1];
  for(int i=0;i<CH;i++){ int idx=begin+i; rowptr[idx]=run; run += cnt_in[idx]; }
  if(t==1023) rowptr[NNODES]=run;
}

__global__ void scatter_kernel(const int* __restrict__ src, const int* __restrict__ dst,
                               const int* __restrict__ rowptr, int* cursor, int* csr){
  int e = blockIdx.x*blockDim.x+threadIdx.x;
  if(e<NEDGES){
    int d = dst[e];
    int p = atomicAdd(&cursor[d],1);
    csr[rowptr[d]+p] = src[e];
  }
}

// ---------------- WMMA f32 GEMM: out[M,128] = act(A[M,K] @ W[K,128] + b) (* scale[row]) ----------
// Block = 256 threads (8 waves). Each wave computes a 16-row x 128-col strip with
// eight 16x16 f32 accumulators driven by v_wmma_f32_16x16x4_f32. W staged in LDS.
template<int K, bool RELU, bool SCALE>
__global__ __launch_bounds__(256) void gemm_wmma(const float* __restrict__ A,
                                                 const float* __restrict__ W,
                                                 const float* __restrict__ bias,
                                                 const float* __restrict__ scale,
                                                 float* __restrict__ out){
  __shared__ float Wsh[K*128];
  int tid = threadIdx.x;
  for(int i=tid;i<K*128;i+=256) Wsh[i]=W[i];
  __syncthreads();

  int lane = tid & 31;
  int wave = tid >> 5;
  int half = lane >> 4;      // which 16-lane half
  int l16  = lane & 15;
  int rowA = blockIdx.x*128 + wave*16 + l16;   // A fragment row (lanes 0-15 <-> M)

  v8f acc[8];
  for(int t=0;t<8;t++) for(int j=0;j<8;j++) acc[t][j]=0.f;

  for(int k=0;k<K;k+=4){
    int ks = k + half*2;     // half 0 -> K,K+1 ; half 1 -> K+2,K+3 (ISA 32-bit A layout)
    v2f a; a[0] = A[rowA*K+ks]; a[1] = A[rowA*K+ks+1];
    for(int t=0;t<8;t++){
      int col = t*16 + l16;
      v2f b; b[0] = Wsh[ks*128 + col]; b[1] = Wsh[(ks+1)*128 + col];
      acc[t] = __builtin_amdgcn_wmma_f32_16x16x4_f32(false, a, false, b,
                                                     (short)0, acc[t], false, false);
    }
  }

  // C/D layout: VGPR j -> M = j + 8*half ; col = lane&15
  int rowBase = blockIdx.x*128 + wave*16 + half*8;
  for(int t=0;t<8;t++){
    int col = t*16 + l16;
    float bv = bias[col];
    for(int j=0;j<8;j++){
      int row = rowBase + j;
      float v = acc[t][j] + bv;
      if(RELU) v = v>0.f ? v : 0.f;
      if(SCALE) v *= scale[row];
      out[row*128 + col] = v;
    }
  }
}

// ---------------- SpMM gather: m[n] = norm_in[n] * sum_{s in in(n)} hs[s] ----------------
// One wave per node; lane owns 4 contiguous floats (float4, 16B aligned). No atomics.
__global__ __launch_bounds__(256) void spmm_kernel(const int* __restrict__ rowptr,
                                                   const int* __restrict__ csr,
                                                   const float* __restrict__ hs,
                                                   const float* __restrict__ norm_in,
                                                   float* __restrict__ m){
  int lane = threadIdx.x & 31;
  int node = blockIdx.x*8 + (threadIdx.x>>5);
  int beg = rowptr[node], end = rowptr[node+1];
  int base = lane*4;
  float a0=0.f,a1=0.f,a2=0.f,a3=0.f;
  for(int e=beg;e<end;e++){
    int s = csr[e];
    float4 v = *(const float4*)(hs + s*128 + base);
    a0+=v.x; a1+=v.y; a2+=v.z; a3+=v.w;
  }
  float ni = norm_in[node];
  float4 o; o.x=a0*ni; o.y=a1*ni; o.z=a2*ni; o.w=a3*ni;
  *(float4*)(m + node*128 + base) = o;
}

// ---------------- LSTM cell: one block per batch row, 128 threads ----------------
__device__ __forceinline__ float sigmoidf_(float x){ return 1.f/(1.f+expf(-x)); }

__global__ __launch_bounds__(128) void lstm_cell(const float* __restrict__ x, int K,
                                                 const float* __restrict__ wih,
                                                 const float* __restrict__ whh,
                                                 const float* __restrict__ bi,
                                                 const float* __restrict__ bh,
                                                 float* __restrict__ h, float* __restrict__ c){
  __shared__ float xs[256];
  __shared__ float hsold[128];
  int b = blockIdx.x, j = threadIdx.x;
  for(int k=j;k<K;k+=128) xs[k]=x[b*K+k];
  hsold[j] = h[b*128+j];
  __syncthreads();
  float gi = bi[j]       + bh[j];
  float gf = bi[128+j]   + bh[128+j];
  float gg = bi[256+j]   + bh[256+j];
  float go = bi[384+j]   + bh[384+j];
  for(int k=0;k<K;k++){
    float xv = xs[k];
    gi += xv*wih[(     j)*K+k];
    gf += xv*wih[(128+j)*K+k];
    gg += xv*wih[(256+j)*K+k];
    go += xv*wih[(384+j)*K+k];
  }
  for(int k=0;k<128;k++){
    float hv = hsold[k];
    gi += hv*whh[(     j)*128+k];
    gf += hv*whh[(128+j)*128+k];
    gg += hv*whh[(256+j)*128+k];
    go += hv*whh[(384+j)*128+k];
  }
  float cn = sigmoidf_(gf)*c[b*128+j] + sigmoidf_(gi)*tanhf(gg);
  float hn = sigmoidf_(go)*tanhf(cn);
  c[b*128+j]=cn; h[b*128+j]=hn;
}

// ---------------- Set2Set attention: one block per graph, 320 threads (1 node each) -----
__global__ __launch_bounds__(320) void attention_kernel(const float* __restrict__ h,
                                                        const float* __restrict__ q,
                                                        float* __restrict__ q_star){
  __shared__ float qs[128];
  __shared__ float esh[PER];
  __shared__ float red[2];
  int g = blockIdx.x, t = threadIdx.x;
  if(t<128) qs[t] = q[g*128+t];
  __syncthreads();
  int n = g*PER + t;
  float e = 0.f;
  for(int d=0; d<128; d++) e += h[n*128+d]*qs[d];
  esh[t]=e;
  __syncthreads();
  if(t==0){ float mx=esh[0]; for(int i=1;i<PER;i++) mx=fmaxf(mx,esh[i]); red[0]=mx; }
  __syncthreads();
  esh[t] = expf(esh[t]-red[0]);
  __syncthreads();
  if(t==0){ float z=0.f; for(int i=0;i<PER;i++) z+=esh[i]; red[1]=z; }
  __syncthreads();
  if(t<128){
    float acc=0.f;
    for(int i=0;i<PER;i++) acc += esh[i]*h[(g*PER+i)*128+t];
    q_star[g*256+t]       = qs[t];
    q_star[g*256+128+t]   = acc/red[1];
  }
}

// ---------------- host-side orchestration ----------------
extern "C" void kernel_launch(void* const* d_in, const int* in_sizes, int n_in,
                              void* d_out, int out_size, void* d_ws, size_t ws_size,
                              hipStream_t stream) {
  const float* feats   = (const float*)d_in[0];
  const int*   src     = (const int*)  d_in[1];
  const int*   dst     = (const int*)  d_in[2];
  const float* embed_w = (const float*)d_in[3];
  const float* embed_b = (const float*)d_in[4];
  const float* conv_w  = (const float*)d_in[5];
  const float* conv_b  = (const float*)d_in[6];
  const float* wih0    = (const float*)d_in[7];
  const float* wih12   = (const float*)d_in[8];
  const float* whh     = (const float*)d_in[9];
  const float* bih     = (const float*)d_in[10];
  const float* bhh     = (const float*)d_in[11];
  float* q_star = (float*)d_out;                       // [128, 256], doubles as Set2Set state

  // workspace carve-out (~66 MB): h matrices are 21 MB each and L2-resident on MI455X (192 MB L2)
  char*  ws  = (char*)d_ws;
  size_t off = 0;
  auto carve = [&](size_t bytes)->char*{ char* p = ws + off; off += (bytes + 255) & ~(size_t)255; return p; };
  int*   cnt_out = (int*)  carve(NNODES*4);
  int*   cnt_in  = (int*)  carve(NNODES*4);
  int*   cursor  = (int*)  carve(NNODES*4);
  int*   rowptr  = (int*)  carve((NNODES+1)*4);
  int*   csr     = (int*)  carve(NEDGES*4);
  float* norm_out= (float*)carve(NNODES*4);
  float* norm_in = (float*)carve(NNODES*4);
  float* hs      = (float*)carve((size_t)NNODES*128*4);   // norm_out-scaled features (gather src)
  float* mbuf    = (float*)carve((size_t)NNODES*128*4);   // aggregated messages
  float* hfin    = (float*)carve((size_t)NNODES*128*4);   // final node embeddings
  float* lh      = (float*)carve(3*NGRAPH*HID*4);
  float* lc      = (float*)carve(3*NGRAPH*HID*4);

  // ---- graph structure: degrees, norms, CSR ----
  zero_i32<<<(NNODES+255)/256,256,0,stream>>>(cnt_out, NNODES);
  zero_i32<<<(NNODES+255)/256,256,0,stream>>>(cnt_in,  NNODES);
  zero_i32<<<(NNODES+255)/256,256,0,stream>>>(cursor,  NNODES);
  degree_count<<<NEDGES/256,256,0,stream>>>(src, dst, cnt_out, cnt_in);
  norm_kernel<<<(NNODES+255)/256,256,0,stream>>>(cnt_out, cnt_in, norm_out, norm_in);
  scan_kernel<<<1,1024,0,stream>>>(cnt_in, rowptr);
  scatter_kernel<<<NEDGES/256,256,0,stream>>>(src, dst, rowptr, cursor, csr);

  // ---- embedding (no relu), fused * norm_out -> hs ----
  gemm_wmma<FIN,false,true><<<NNODES/128,256,0,stream>>>(feats, embed_w, embed_b, norm_out, hs);

  // ---- 4 GraphConv layers: gather-SpMM then WMMA GEMM (+bias, relu, fused scaling) ----
  for(int l=0;l<4;l++){
    spmm_kernel<<<NNODES/8,256,0,stream>>>(rowptr, csr, hs, norm_in, mbuf);
    const float* Wl = conv_w + (size_t)l*128*128;
    const float* bl = conv_b + l*128;
    if(l<3) gemm_wmma<HID,true,true ><<<NNODES/128,256,0,stream>>>(mbuf, Wl, bl, norm_out, hs);
    else    gemm_wmma<HID,true,false><<<NNODES/128,256,0,stream>>>(mbuf, Wl, bl, nullptr,  hfin);
  }

  // ---- Set2Set readout ----
  zero_f32<<<(3*NGRAPH*HID+255)/256,256,0,stream>>>(lh, 3*NGRAPH*HID);
  zero_f32<<<(3*NGRAPH*HID+255)/256,256,0,stream>>>(lc, 3*NGRAPH*HID);
  zero_f32<<<(NGRAPH*2*HID+255)/256,256,0,stream>>>(q_star, NGRAPH*2*HID);

  for(int it=0; it<6; ++it){
    for(int layer=0; layer<3; ++layer){
      const float* x; int K; const float* wih;
      if(layer==0){ x = q_star; K = 2*HID; wih = wih0; }
      else        { x = lh + (layer-1)*NGRAPH*HID; K = HID; wih = wih12 + (size_t)(layer-1)*512*HID; }
      lstm_cell<<<NGRAPH,128,0,stream>>>(x, K, wih,
                                         whh + (size_t)layer*512*HID,
                                         bih + layer*512, bhh + layer*512,
                                         lh + layer*NGRAPH*HID, lc + layer*NGRAPH*HID);
    }
    attention_kernel<<<NGRAPH,PER,0,stream>>>(hfin, lh + 2*NGRAPH*HID, q_star);
  }
  (void)in_sizes; (void)n_in; (void)out_size; (void)ws_size;
}